// TopKAbsolutes1D_27504970563633
// MI455X (gfx1250) — compile-verified
//
#include <hip/hip_runtime.h>
#include <stdint.h>

// TopKAbsolutes1D for MI455X (gfx1250):
//   input  (64,64,16384) fp32, topk=128 along last axis.
//   One workgroup per row. Row staged into LDS via the Tensor Data Mover
//   (single tensor_load_to_lds per row), radix-256 select on |x| bit
//   patterns entirely in LDS, one coalesced write pass back to HBM.
//   HBM traffic = 1 read + 1 write of the tensor (~512 MiB -> ~22 us floor).

#define ROW_LEN   16384
#define NTHREADS  256
#define TIE_CAP   256

typedef unsigned int u32;
typedef unsigned int u32x4 __attribute__((ext_vector_type(4)));
typedef int          i32x8 __attribute__((ext_vector_type(8)));
typedef int          i32x4 __attribute__((ext_vector_type(4)));

__global__ __launch_bounds__(NTHREADS)
void topk_abs_rowselect_kernel(const u32* __restrict__ x,
                               const int* __restrict__ kptr,
                               u32* __restrict__ out)
{
    __shared__ u32 row[ROW_LEN];     // 64 KB: raw fp32 bit patterns of this row
    __shared__ u32 hist[256];        // radix histogram
    __shared__ u32 tieIdx[TIE_CAP];  // indices of |x| == threshold
    __shared__ u32 sc[3];            // 0: prefix, 1: remaining_k, 2: tie count

    const int tid = (int)threadIdx.x;
    const u32 rowId = blockIdx.x;
    const u32* __restrict__ src = x + (size_t)rowId * ROW_LEN;
    u32* __restrict__ dst = out + (size_t)rowId * ROW_LEN;

    if (tid == 0) sc[2] = 0u;

    // ---------------- Stage row into LDS ----------------
#if defined(__gfx1250__) && __has_builtin(__builtin_amdgcn_tensor_load_to_lds) && \
    __has_builtin(__builtin_amdgcn_s_wait_tensorcnt)
    {
        // Only wave 0 issues the TDM op (TDM ignores EXEC; a predicated issue
        // from all 8 waves would launch 8 duplicate DMAs). readfirstlane makes
        // the wave-id test a uniform scalar branch.
        const int waveId = __builtin_amdgcn_readfirstlane(tid >> 5);
        if (waveId == 0) {
            const u32 ldsAddr = (u32)(uintptr_t)(&row[0]); // low 32 bits = LDS byte offset
            const uint64_t ga = (uint64_t)(uintptr_t)src;

            // D# group 0 (128b): count=1 | lds_addr[63:32] | global_addr[120:64] | type=2
            u32x4 g0;
            g0.x = 1u;                                   // count=1, user descriptor
            g0.y = ldsAddr;                              // LDS byte address
            g0.z = (u32)(ga & 0xffffffffu);              // global_addr[31:0]
            g0.w = (u32)((ga >> 32) & 0x01ffffffu)       // global_addr[56:32]
                 | (2u << 30);                           // type = 2 ("image")

            // D# group 1 (256b): 2-D tensor/tile, 4-byte elements, no multicast.
            i32x8 g1;
            g1[0] = (int)(2u << 16);                     // workgroup_mask=0, data_size=2 (4B)
            g1[1] = (int)(((u32)ROW_LEN & 0xffffu) << 16);        // tensor_dim0[15:0] @ [63:48]
            g1[2] = (int)((((u32)ROW_LEN >> 16) & 0xffffu)        // tensor_dim0[31:16]
                 | (1u << 16));                                   // tensor_dim1 = 1 (lo16)
            g1[3] = (int)(((u32)ROW_LEN & 0xffffu) << 16);        // tensor_dim1 hi=0 | tile_dim0=16384
            g1[4] = (int)1;                              // tile_dim1 = 1, tile_dim2 = 0 (unused)
            g1[5] = (int)ROW_LEN;                        // tensor_dim0_stride[31:0] (elements)
            g1[6] = 0;                                   // stride0 hi16 = 0, stride1 lo16 = 0
            g1[7] = 0;                                   // stride1 hi = 0

            i32x4 z4 = {0, 0, 0, 0};                     // groups 2/3 unused (<=2-D tensor)
#if __clang_major__ >= 23
            i32x8 z8 = {0, 0, 0, 0, 0, 0, 0, 0};
            __builtin_amdgcn_tensor_load_to_lds(g0, g1, z4, z4, z8, 0);
#else
            __builtin_amdgcn_tensor_load_to_lds(g0, g1, z4, z4, 0);
#endif
            __builtin_amdgcn_s_wait_tensorcnt(0);        // s_wait_tensorcnt 0
        }
    }
    __syncthreads();   // publish TDM-written LDS to all 8 waves
#else
    // Fallback: vectorized cooperative copy global -> LDS.
    for (int i = tid * 4; i < ROW_LEN; i += NTHREADS * 4) {
        const uint4 v = *reinterpret_cast<const uint4*>(src + i);
        *reinterpret_cast<uint4*>(&row[i]) = v;
    }
    __syncthreads();
#endif

    const u32 k = (u32)(*kptr);   // 128

    // ---------------- Radix-256 MSD select on |x| bit pattern ----------------
    // For non-negative IEEE floats, unsigned-int compare == float compare, so
    // the k-th largest |x| can be found byte-by-byte from the MSB.
    u32 prefix = 0u;   // high bytes of the threshold fixed so far
    u32 remk   = k;    // how many of the participating elements we still keep
    for (int d = 3; d >= 0; --d) {
        for (int b = tid; b < 256; b += NTHREADS) hist[b] = 0u;
        __syncthreads();

        const int sh = 8 * d;
        const uint64_t hiPrefix = (uint64_t)prefix >> (sh + 8);   // 64-bit: sh+8 can be 32
        for (int i = tid; i < ROW_LEN; i += NTHREADS) {
            const u32 a = row[i] & 0x7fffffffu;
            if (((uint64_t)a >> (sh + 8)) == hiPrefix) {
                atomicAdd(&hist[(a >> sh) & 0xffu], 1u);
            }
        }
        __syncthreads();

        if (tid == 0) {
            u32 c = 0u;
            int b = 255;
            for (; b > 0; --b) {
                const u32 h = hist[b];
                if (c + h >= remk) break;
                c += h;
            }
            sc[0] = prefix | ((u32)b << sh);
            sc[1] = remk - c;
        }
        __syncthreads();
        prefix = sc[0];
        remk   = sc[1];
    }

    const u32 T = prefix;      // exact bit pattern of the k-th largest |x|
    const u32 rkeep = remk;    // number of |x|==T elements to keep (usually 1)

    // ---------------- Output pass (LDS -> HBM) ----------------
    for (int i = tid; i < ROW_LEN; i += NTHREADS) {
        const u32 v = row[i];
        const u32 a = v & 0x7fffffffu;
        if (a > T) {
            dst[i] = v;                                // keeper, signed value
        } else if (a < T) {
            dst[i] = 0u;                               // zeroed
        } else {
            const u32 slot = atomicAdd(&sc[2], 1u);    // tie at threshold
            if (slot < TIE_CAP) tieIdx[slot] = (u32)i;
            else                dst[i] = 0u;           // pathological overflow: drop
        }
    }
    __syncthreads();

    // Tie fixup: keep the rkeep ties with smallest index (top_k tie order).
    u32 m = sc[2];
    if (m > TIE_CAP) m = TIE_CAP;
    for (u32 j = (u32)tid; j < m; j += NTHREADS) {
        const u32 i = tieIdx[j];
        u32 rank = 0u;
        for (u32 jj = 0; jj < m; ++jj) rank += (tieIdx[jj] < i) ? 1u : 0u;
        dst[i] = (rank < rkeep) ? row[i] : 0u;
    }
}

extern "C" void kernel_launch(void* const* d_in, const int* in_sizes, int n_in,
                              void* d_out, int out_size, void* d_ws, size_t ws_size,
                              hipStream_t stream) {
    (void)n_in; (void)d_ws; (void)ws_size; (void)out_size;
    const u32* x = (const u32*)d_in[0];         // fp32 bit patterns
    const int* kptr = (const int*)d_in[1];      // topk (single-element array)
    u32* out = (u32*)d_out;                     // fp32 output, written as bits

    const int rows = in_sizes[0] / ROW_LEN;     // 64*64 = 4096
    topk_abs_rowselect_kernel<<<dim3(rows), dim3(NTHREADS), 0, stream>>>(x, kptr, out);
}